// GlobalUpdate_39041252721121
// MI455X (gfx1250) — compile-verified
//
#include <hip/hip_runtime.h>
#include <hip/hip_bf16.h>

typedef float v2f __attribute__((ext_vector_type(2)));
typedef float v4f __attribute__((ext_vector_type(4)));
typedef float v8f __attribute__((ext_vector_type(8)));

#define NNODES 500000
#define BSEG   4096
#define DS_    128
#define VROW   375     // 3 * 125
#define NGRP   126     // 32 float4 groups (x_s) + 94 groups (x_v, last has 3 elems)

// ---------------------------------------------------------------- zero scratch
__global__ void zero_ws_kernel(float* ws, long long n) {
  long long idx = (long long)blockIdx.x * blockDim.x + threadIdx.x;
  long long stride = (long long)gridDim.x * blockDim.x;
  for (; idx < n; idx += stride) ws[idx] = 0.0f;
}

// --------------------------------------------------- segment-sum via L2 atomics
// ~1 GB of streaming reads (non-temporal, b128) -> f32 atomics into an 8 MB
// accumulator that stays resident in the 192 MB L2.
__global__ void seg_accum_kernel(const float* __restrict__ xs,
                                 const float* __restrict__ xv,
                                 const int*   __restrict__ seg,
                                 float* __restrict__ cnt,
                                 float* __restrict__ sum_s,
                                 float* __restrict__ sum_v) {
  const long long total = (long long)NNODES * NGRP;
  long long idx = (long long)blockIdx.x * blockDim.x + threadIdx.x;
  const long long stride = (long long)gridDim.x * blockDim.x;
  for (; idx < total; idx += stride) {
    int node = (int)(idx / NGRP);
    int g    = (int)(idx % NGRP);
    int s = seg[node];
    if (g < 32) {
      int f = g * 4;
      v4f v = __builtin_nontemporal_load(
          (const v4f*)(xs + (long long)node * DS_ + f));
      float* d = sum_s + (long long)s * DS_ + f;
      atomicAdd(d + 0, v.x);
      atomicAdd(d + 1, v.y);
      atomicAdd(d + 2, v.z);
      atomicAdd(d + 3, v.w);
      if (g == 0) atomicAdd(&cnt[s], 1.0f);
    } else {
      int c = (g - 32) * 4;
      const float* p = xv + (long long)node * VROW + c;
      float* d = sum_v + (long long)s * VROW + c;
      if (c < 372) {
        v4f v = __builtin_nontemporal_load((const v4f*)p);
        atomicAdd(d + 0, v.x);
        atomicAdd(d + 1, v.y);
        atomicAdd(d + 2, v.z);
        atomicAdd(d + 3, v.w);
      } else {                       // tail: features 372..374
        atomicAdd(d + 0, __builtin_nontemporal_load(p + 0));
        atomicAdd(d + 1, __builtin_nontemporal_load(p + 1));
        atomicAdd(d + 2, __builtin_nontemporal_load(p + 2));
      }
    }
  }
}

// ------------------------------------------------------------- GVP finalize
// One wave32 per 16 graphs. All large matmuls via V_WMMA_F32_16X16X4_F32.
__global__ __launch_bounds__(32)
void gvp_finalize_kernel(const float* __restrict__ cnt,
                         const float* __restrict__ sum_s,
                         const float* __restrict__ sum_v,
                         const float* __restrict__ u_s,
                         const float* __restrict__ u_v,
                         const float* __restrict__ Wd,  const float* __restrict__ bd,
                         const float* __restrict__ W1h, const float* __restrict__ W1vo,
                         const float* __restrict__ W1so,const float* __restrict__ b1so,
                         const float* __restrict__ W1g, const float* __restrict__ b1g,
                         const float* __restrict__ W2h, const float* __restrict__ W2vo,
                         const float* __restrict__ W2so,const float* __restrict__ b2so,
                         const float* __restrict__ W2g, const float* __restrict__ b2g,
                         float* __restrict__ out_s2, float* __restrict__ out_v2) {
  __shared__ float sm_s[16][128];   // mean_s  -> later sh -> later sh2 (cols 0..63)
  __shared__ float sm_v[48][128];   // update_v (rows b*3+x) -> later V1 (cols 0..63)
  __shared__ float sm_vh[48][128];  // Vh -> later Vh2 (cols 0..63)
  __shared__ float sm_us[16][32];   // update_s = [avg_s@Wd+bd | u_s]
  __shared__ float sm_so[16][12];   // s_out (10 used)
  __shared__ float sm_g1[16][64];   // gate1
  __shared__ float sm_vv2[48][4];   // Vvo2 (3 used)
  __shared__ float inv_cnt[16];

  const int lane = threadIdx.x;
  const int half = lane >> 4;       // 0 or 1
  const int m    = lane & 15;
  const int g0   = blockIdx.x * 16;

  if (lane < 16) {
    float c = cnt[g0 + lane];
    inv_cnt[lane] = 1.0f / fmaxf(c, 1.0f);
  }
  __syncthreads();

  // ---- stage means / concat inputs into LDS
  for (int e = lane; e < 16 * 128; e += 32) {
    int b = e >> 7, f = e & 127;
    sm_s[b][f] = sum_s[(long long)(g0 + b) * 128 + f] * inv_cnt[b];
  }
  for (int e = lane; e < 48 * 128; e += 32) {
    int r = e >> 7, c = e & 127;
    int b = r / 3, x = r % 3;
    float v;
    if (c < 125) v = sum_v[(long long)(g0 + b) * 375 + x * 125 + c] * inv_cnt[b];
    else         v = u_v[(g0 + b) * 9 + x * 3 + (c - 125)];
    sm_v[r][c] = v;
  }
  for (int e = lane; e < 16 * 16; e += 32) {
    int b = e >> 4, k = e & 15;
    sm_us[b][16 + k] = u_s[(g0 + b) * 16 + k];
  }
  __syncthreads();

  // ---- avg_s @ Wd  ([16,128] x [128,16]) : 32 WMMA K-steps
  {
    v8f acc = {};
    for (int kk = 0; kk < 32; ++kk) {
      int k0 = kk * 4;
      v2f a, bb;
      a.x  = sm_s[m][k0 + 2 * half];
      a.y  = sm_s[m][k0 + 2 * half + 1];
      bb.x = Wd[(k0 + 2 * half) * 16 + m];
      bb.y = Wd[(k0 + 2 * half + 1) * 16 + m];
      acc = __builtin_amdgcn_wmma_f32_16x16x4_f32(false, a, false, bb,
                                                  (short)0, acc, false, false);
    }
    for (int j = 0; j < 8; ++j) {
      int M = j + 8 * half;
      sm_us[M][m] = acc[j] + bd[m];
    }
  }
  __syncthreads();

  // ---- Vh = update_v @ W1h  ([48,128] x [128,128])
  for (int t = 0; t < 3; ++t) {
    for (int nt = 0; nt < 8; ++nt) {
      int n0 = nt * 16;
      v8f acc = {};
      for (int kk = 0; kk < 32; ++kk) {
        int k0 = kk * 4;
        v2f a, bb;
        a.x  = sm_v[t * 16 + m][k0 + 2 * half];
        a.y  = sm_v[t * 16 + m][k0 + 2 * half + 1];
        bb.x = W1h[(k0 + 2 * half) * 128 + n0 + m];
        bb.y = W1h[(k0 + 2 * half + 1) * 128 + n0 + m];
        acc = __builtin_amdgcn_wmma_f32_16x16x4_f32(false, a, false, bb,
                                                    (short)0, acc, false, false);
      }
      for (int j = 0; j < 8; ++j)
        sm_vh[t * 16 + j + 8 * half][n0 + m] = acc[j];
    }
  }
  __syncthreads();

  // ---- sh = ||Vh|| over x  (reuse sm_s)
  for (int e = lane; e < 16 * 128; e += 32) {
    int b = e >> 7, h = e & 127;
    int r = b * 3;
    float a0 = sm_vh[r][h], a1 = sm_vh[r + 1][h], a2 = sm_vh[r + 2][h];
    sm_s[b][h] = sqrtf(a0 * a0 + a1 * a1 + a2 * a2);
  }
  __syncthreads();

  // ---- s_out = relu([sh | update_s] @ W1so + b1so)   [16,10]
  if (lane < 16) {
    int b = lane;
    for (int o = 0; o < 10; ++o) {
      float a = b1so[o];
      for (int k = 0; k < 128; ++k) a += sm_s[b][k] * W1so[k * 10 + o];
      for (int k = 0; k < 32; ++k)  a += sm_us[b][k] * W1so[(128 + k) * 10 + o];
      sm_so[b][o] = fmaxf(a, 0.0f);
    }
  }
  __syncthreads();

  // ---- gate1 = sigmoid(sigmoid(s_out) @ W1g + b1g)   [16,64]
  {
    int b = m;
    float vs[10];
    for (int k = 0; k < 10; ++k) vs[k] = 1.0f / (1.0f + __expf(-sm_so[b][k]));
    for (int o = half * 32; o < half * 32 + 32; ++o) {
      float g = b1g[o];
      for (int k = 0; k < 10; ++k) g += vs[k] * W1g[k * 64 + o];
      sm_g1[b][o] = 1.0f / (1.0f + __expf(-g));
    }
  }
  __syncthreads();

  // ---- V1 = (Vh @ W1vo) * gate1   ([48,128]x[128,64], store into sm_v cols 0..63)
  for (int t = 0; t < 3; ++t) {
    for (int nt = 0; nt < 4; ++nt) {
      int n0 = nt * 16;
      v8f acc = {};
      for (int kk = 0; kk < 32; ++kk) {
        int k0 = kk * 4;
        v2f a, bb;
        a.x  = sm_vh[t * 16 + m][k0 + 2 * half];
        a.y  = sm_vh[t * 16 + m][k0 + 2 * half + 1];
        bb.x = W1vo[(k0 + 2 * half) * 64 + n0 + m];
        bb.y = W1vo[(k0 + 2 * half + 1) * 64 + n0 + m];
        acc = __builtin_amdgcn_wmma_f32_16x16x4_f32(false, a, false, bb,
                                                    (short)0, acc, false, false);
      }
      for (int j = 0; j < 8; ++j) {
        int R = t * 16 + j + 8 * half;
        sm_v[R][n0 + m] = acc[j] * sm_g1[R / 3][n0 + m];
      }
    }
  }
  __syncthreads();

  // ---- Vh2 = V1 @ W2h  ([48,64] x [64,64], into sm_vh cols 0..63)
  for (int t = 0; t < 3; ++t) {
    for (int nt = 0; nt < 4; ++nt) {
      int n0 = nt * 16;
      v8f acc = {};
      for (int kk = 0; kk < 16; ++kk) {
        int k0 = kk * 4;
        v2f a, bb;
        a.x  = sm_v[t * 16 + m][k0 + 2 * half];
        a.y  = sm_v[t * 16 + m][k0 + 2 * half + 1];
        bb.x = W2h[(k0 + 2 * half) * 64 + n0 + m];
        bb.y = W2h[(k0 + 2 * half + 1) * 64 + n0 + m];
        acc = __builtin_amdgcn_wmma_f32_16x16x4_f32(false, a, false, bb,
                                                    (short)0, acc, false, false);
      }
      for (int j = 0; j < 8; ++j)
        sm_vh[t * 16 + j + 8 * half][n0 + m] = acc[j];
    }
  }
  __syncthreads();

  // ---- Vvo2 = Vh2 @ W2vo  ([48,64]x[64,3], scalar: tiny)
  for (int e = lane; e < 48 * 3; e += 32) {
    int r = e / 3, c = e % 3;
    float a = 0.0f;
    for (int k = 0; k < 64; ++k) a += sm_vh[r][k] * W2vo[k * 3 + c];
    sm_vv2[r][c] = a;
  }
  // ---- sh2 (reuse sm_s cols 0..63)
  for (int e = lane; e < 16 * 64; e += 32) {
    int b = e >> 6, h = e & 63;
    int r = b * 3;
    float a0 = sm_vh[r][h], a1 = sm_vh[r + 1][h], a2 = sm_vh[r + 2][h];
    sm_s[b][h] = sqrtf(a0 * a0 + a1 * a1 + a2 * a2);
  }
  __syncthreads();

  // ---- heads + outputs
  if (lane < 16) {
    int b = lane;
    float s2[3];
    for (int o = 0; o < 3; ++o) {
      float a = b2so[o];
      for (int k = 0; k < 64; ++k) a += sm_s[b][k] * W2so[k * 3 + o];
      for (int k = 0; k < 10; ++k) a += sm_so[b][k] * W2so[(64 + k) * 3 + o];
      s2[o] = fmaxf(a, 0.0f);
      out_s2[(g0 + b) * 3 + o] = s2[o];
    }
    float vs[3], g2[3];
    for (int k = 0; k < 3; ++k) vs[k] = 1.0f / (1.0f + __expf(-s2[k]));
    for (int o = 0; o < 3; ++o) {
      float g = b2g[o];
      for (int k = 0; k < 3; ++k) g += vs[k] * W2g[k * 3 + o];
      g2[o] = 1.0f / (1.0f + __expf(-g));
    }
    for (int x = 0; x < 3; ++x)
      for (int c = 0; c < 3; ++c)
        out_v2[((g0 + b) * 3 + x) * 3 + c] = sm_vv2[b * 3 + x][c] * g2[c];
  }
}

extern "C" void kernel_launch(void* const* d_in, const int* in_sizes, int n_in,
                              void* d_out, int out_size, void* d_ws, size_t ws_size,
                              hipStream_t stream) {
  const float* x_s  = (const float*)d_in[0];
  const float* x_v  = (const float*)d_in[1];
  const int*   seg  = (const int*)d_in[2];
  const float* u_s  = (const float*)d_in[3];
  const float* u_v  = (const float*)d_in[4];
  const float* Wd   = (const float*)d_in[5];
  const float* bd   = (const float*)d_in[6];
  const float* W1h  = (const float*)d_in[7];
  const float* W1vo = (const float*)d_in[8];
  const float* W1so = (const float*)d_in[9];
  const float* b1so = (const float*)d_in[10];
  const float* W1g  = (const float*)d_in[11];
  const float* b1g  = (const float*)d_in[12];
  const float* W2h  = (const float*)d_in[13];
  const float* W2vo = (const float*)d_in[14];
  const float* W2so = (const float*)d_in[15];
  const float* b2so = (const float*)d_in[16];
  const float* W2g  = (const float*)d_in[17];
  const float* b2g  = (const float*)d_in[18];

  float* ws    = (float*)d_ws;
  float* cnt   = ws;                          // B
  float* sum_s = ws + BSEG;                   // B*128
  float* sum_v = ws + BSEG + BSEG * DS_;      // B*375

  long long zn = (long long)BSEG * (1 + DS_ + VROW);
  int zb = (int)((zn + 255) / 256);
  zero_ws_kernel<<<zb, 256, 0, stream>>>(ws, zn);

  long long total = (long long)NNODES * NGRP;
  long long ab64 = (total + 255) / 256;
  int ab = (ab64 > 262144) ? 262144 : (int)ab64;
  seg_accum_kernel<<<ab, 256, 0, stream>>>(x_s, x_v, seg, cnt, sum_s, sum_v);

  float* out_s2 = (float*)d_out;
  float* out_v2 = out_s2 + (long long)BSEG * 3;
  gvp_finalize_kernel<<<BSEG / 16, 32, 0, stream>>>(
      cnt, sum_s, sum_v, u_s, u_v,
      Wd, bd, W1h, W1vo, W1so, b1so, W1g, b1g,
      W2h, W2vo, W2so, b2so, W2g, b2g,
      out_s2, out_v2);
}